// Decoder_33990371181267
// MI455X (gfx1250) — compile-verified
//
#include <hip/hip_runtime.h>
#include <hip/hip_bf16.h>

#define H   128
#define LC  32
#define KNN_K 16
#define KNN_BLOCK 128
#define KNN_CHUNK 1024

typedef float v2f __attribute__((ext_vector_type(2)));
typedef float v4f __attribute__((ext_vector_type(4)));
typedef float v8f __attribute__((ext_vector_type(8)));

__device__ __forceinline__ float elu_f(float x) {
    return x > 0.0f ? x : expm1f(x);
}

// ---------------------------------------------------------------------------
// lin_1: out[n,c] = elu(sum_k latent[n,k] * w[k,c] + b[c]),  K = 32
// ---------------------------------------------------------------------------
__global__ void lin1_elu_kernel(const float* __restrict__ latent,
                                const float* __restrict__ w,
                                const float* __restrict__ b,
                                float* __restrict__ out, int n)
{
    int i = blockIdx.x * blockDim.x + threadIdx.x;
    if (i >= n * H) return;
    int node = i >> 7;
    int c    = i & (H - 1);
    float acc = b[c];
    const float* lrow = latent + (size_t)node * LC;
#pragma unroll
    for (int k = 0; k < LC; ++k)
        acc += lrow[k] * w[k * H + c];
    out[i] = elu_f(acc);
}

// ---------------------------------------------------------------------------
// float4 copy (y = x) : seeds y with the GIN self term before edge atomics
// ---------------------------------------------------------------------------
__global__ void copy_f4_kernel(const float* __restrict__ a,
                               float* __restrict__ b, long n4)
{
    long i = (long)blockIdx.x * blockDim.x + threadIdx.x;
    if (i < n4) ((v4f*)b)[i] = ((const v4f*)a)[i];
}

// ---------------------------------------------------------------------------
// Edge aggregation: y[dst] += x[src]  (segment_sum).  One thread per
// (edge, 4-float chunk): coalesced 512B row reads, f32 atomics resolve in L2.
// ---------------------------------------------------------------------------
__global__ void edge_agg_kernel(const int* __restrict__ src,
                                const int* __restrict__ dst,
                                const float* __restrict__ x,
                                float* __restrict__ y, int E)
{
    long i = (long)blockIdx.x * blockDim.x + threadIdx.x;
    long total = (long)E * (H / 4);
    if (i >= total) return;
    int e = (int)(i >> 5);
    int c = (int)(i & 31);
    int s = src[e];
    int d = dst[e];
    v4f v = *(const v4f*)(x + (size_t)s * H + c * 4);
    float* yd = y + (size_t)d * H + c * 4;
    atomicAdd(yd + 0, v.x);
    atomicAdd(yd + 1, v.y);
    atomicAdd(yd + 2, v.z);
    atomicAdd(yd + 3, v.w);
}

// ---------------------------------------------------------------------------
// In-place 128x128 GEMM + bias (+ optional ELU) using V_WMMA_F32_16X16X4_F32.
//   x[n,128] <- act(x[n,128] @ W[128,128] + bias)
// 256 threads = 8 waves; each wave owns a 16-row x 128-col strip (8 v8f accs).
// W is repacked once per block into LDS in B-fragment order
// ([kstep][coltile][lane] float2) so B reads are conflict-free ds_load_b64.
// A-fragment layout (16x4 f32): lanes 0-15 hold K=4s+0/1, lanes 16-31 K=4s+2/3.
// C/D layout: VGPR v -> rows v (lanes 0-15) and v+8 (lanes 16-31).
// ---------------------------------------------------------------------------
__global__ __launch_bounds__(256)
void gemm_h128_bias_act_kernel(float* __restrict__ x,
                               const float* __restrict__ W,
                               const float* __restrict__ bias,
                               int n, int act)
{
    __shared__ float wl[32 * 8 * 32 * 2];   // 16384 floats = 64 KB

    const int tid = threadIdx.x;

    // Cooperative repack of W into fragment order (8192 float2 entries).
    for (int i = tid; i < 8192; i += 256) {
        int lane = i & 31;
        int ct   = (i >> 5) & 7;
        int s    = i >> 8;                  // k-step 0..31
        int hi   = lane >> 4;
        int col  = ct * 16 + (lane & 15);
        int k0   = s * 4 + hi * 2;
        wl[i * 2 + 0] = W[(k0 + 0) * H + col];
        wl[i * 2 + 1] = W[(k0 + 1) * H + col];
    }
    __syncthreads();

    const int wave = tid >> 5;
    const int lane = tid & 31;
    const int hi   = lane >> 4;
    const int r0   = blockIdx.x * 128 + wave * 16;

    // A-fragment source row for this lane (clamped so EXEC stays all-ones).
    int aRow = r0 + (lane & 15);
    if (aRow > n - 1) aRow = n - 1;
    const float* arow = x + (size_t)aRow * H;

    v8f acc[8];
#pragma unroll
    for (int ct = 0; ct < 8; ++ct) acc[ct] = (v8f)0.0f;

    for (int s = 0; s < 32; ++s) {
        v2f a = *(const v2f*)(arow + s * 4 + hi * 2);
#pragma unroll
        for (int ct = 0; ct < 8; ++ct) {
            v2f b = *(const v2f*)(&wl[((s * 8 + ct) * 32 + lane) * 2]);
            acc[ct] = __builtin_amdgcn_wmma_f32_16x16x4_f32(
                false, a, false, b, (short)0, acc[ct], false, false);
        }
    }

    // Epilogue: bias + optional ELU, guarded in-place store.
#pragma unroll
    for (int ct = 0; ct < 8; ++ct) {
        int col = ct * 16 + (lane & 15);
        float bv = bias[col];
#pragma unroll
        for (int v = 0; v < 8; ++v) {
            int row = r0 + v + hi * 8;
            if (row < n) {
                float r = acc[ct][v] + bv;
                if (act) r = elu_f(r);
                x[(size_t)row * H + col] = r;
            }
        }
    }
}

// ---------------------------------------------------------------------------
// Final Linear: out[n,3] = h[n,128] @ wL1[128,3] + bL1   (tiny, VALU)
// ---------------------------------------------------------------------------
__global__ void gemm_out3_kernel(const float* __restrict__ h,
                                 const float* __restrict__ w,
                                 const float* __restrict__ b,
                                 float* __restrict__ out, int n)
{
    int t = blockIdx.x * blockDim.x + threadIdx.x;
    if (t >= n) return;
    float o0 = b[0], o1 = b[1], o2 = b[2];
    const float* row = h + (size_t)t * H;
#pragma unroll 4
    for (int k = 0; k < H; ++k) {
        float v = row[k];
        o0 += v * w[k * 3 + 0];
        o1 += v * w[k * 3 + 1];
        o2 += v * w[k * 3 + 2];
    }
    out[t * 3 + 0] = o0;
    out[t * 3 + 1] = o1;
    out[t * 3 + 2] = o2;
}

// ---------------------------------------------------------------------------
// kNN top-16 (smallest squared distance). One thread per target; source
// positions staged through LDS in chunks; per-thread 16-best list kept in
// LDS with [slot][tid] layout (stride 128 => bank = tid mod 64, conflict-free
// for any data-dependent slot index).
// ---------------------------------------------------------------------------
__global__ __launch_bounds__(KNN_BLOCK)
void knn_topk_kernel(const float* __restrict__ pos_src,
                     const float* __restrict__ pos_tgt,
                     int ns, int nt,
                     int* __restrict__ out_idx,
                     float* __restrict__ out_w)
{
    __shared__ float sx[KNN_CHUNK], sy[KNN_CHUNK], sz[KNN_CHUNK];
    __shared__ float bd[KNN_K][KNN_BLOCK];
    __shared__ int   bi[KNN_K][KNN_BLOCK];

    const int tid = threadIdx.x;
    const int t   = blockIdx.x * KNN_BLOCK + tid;

    float px = 0.f, py = 0.f, pz = 0.f;
    if (t < nt) {
        px = pos_tgt[(size_t)t * 3 + 0];
        py = pos_tgt[(size_t)t * 3 + 1];
        pz = pos_tgt[(size_t)t * 3 + 2];
    }
#pragma unroll
    for (int j = 0; j < KNN_K; ++j) { bd[j][tid] = 3.4e38f; bi[j][tid] = 0; }
    float worst = 3.4e38f;
    int   wslot = 0;

    for (int base = 0; base < ns; base += KNN_CHUNK) {
        int cnt = ns - base; if (cnt > KNN_CHUNK) cnt = KNN_CHUNK;
        __syncthreads();
        for (int i = tid; i < cnt; i += KNN_BLOCK) {
            sx[i] = pos_src[(size_t)(base + i) * 3 + 0];
            sy[i] = pos_src[(size_t)(base + i) * 3 + 1];
            sz[i] = pos_src[(size_t)(base + i) * 3 + 2];
        }
        __syncthreads();
        for (int i = 0; i < cnt; ++i) {
            float dx = px - sx[i];
            float dy = py - sy[i];
            float dz = pz - sz[i];
            float d2 = dx * dx + dy * dy + dz * dz;
            if (d2 < worst) {
                bd[wslot][tid] = d2;
                bi[wslot][tid] = base + i;
                worst = -1.0f;
#pragma unroll
                for (int j = 0; j < KNN_K; ++j) {
                    float v = bd[j][tid];
                    if (v > worst) { worst = v; wslot = j; }
                }
            }
        }
    }

    if (t < nt) {
#pragma unroll
        for (int j = 0; j < KNN_K; ++j) {
            float d2 = bd[j][tid];
            out_w[(size_t)t * KNN_K + j]   = 1.0f / fmaxf(d2, 1e-16f);
            out_idx[(size_t)t * KNN_K + j] = bi[j][tid];
        }
    }
}

// ---------------------------------------------------------------------------
// kNN gather: out[t] = sum_j w_j * x[idx_j] / sum_j w_j.
// One wave per target; lane handles 4 features -> 512B coalesced row reads.
// ---------------------------------------------------------------------------
__global__ __launch_bounds__(128)
void knn_gather_kernel(const float* __restrict__ xsrc,
                       const int* __restrict__ idx,
                       const float* __restrict__ w,
                       float* __restrict__ xdst, int nt)
{
    int wave = threadIdx.x >> 5;
    int lane = threadIdx.x & 31;
    int t = blockIdx.x * 4 + wave;
    if (t >= nt) return;

    v4f acc = (v4f)0.0f;
    float wsum = 0.0f;
#pragma unroll
    for (int j = 0; j < KNN_K; ++j) {
        float wj = w[(size_t)t * KNN_K + j];
        int   sj = idx[(size_t)t * KNN_K + j];
        v4f row = *(const v4f*)(xsrc + (size_t)sj * H + lane * 4);
        acc += wj * row;
        wsum += wj;
    }
    acc *= (1.0f / wsum);
    *(v4f*)(xdst + (size_t)t * H + lane * 4) = acc;
}

// ---------------------------------------------------------------------------
// Host orchestration
// ---------------------------------------------------------------------------
static void run_gin(float* xin, float* y, const int* ei, int E, int n,
                    const float* w0, const float* b0,
                    const float* w1, const float* b1,
                    int outer_act, hipStream_t stream)
{
    long n4 = (long)n * H / 4;
    copy_f4_kernel<<<(unsigned)((n4 + 255) / 256), 256, 0, stream>>>(xin, y, n4);
    long tw = (long)E * (H / 4);
    edge_agg_kernel<<<(unsigned)((tw + 255) / 256), 256, 0, stream>>>(
        ei, ei + E, xin, y, E);
    int gb = (n + 127) / 128;
    gemm_h128_bias_act_kernel<<<gb, 256, 0, stream>>>(y, w0, b0, n, 1);
    gemm_h128_bias_act_kernel<<<gb, 256, 0, stream>>>(y, w1, b1, n, outer_act);
}

extern "C" void kernel_launch(void* const* d_in, const int* in_sizes, int n_in,
                              void* d_out, int out_size, void* d_ws, size_t ws_size,
                              hipStream_t stream)
{
    const float* latent = (const float*)d_in[0];
    const float* pos0   = (const float*)d_in[1];
    const float* pos1   = (const float*)d_in[2];
    const float* pos2   = (const float*)d_in[3];
    const int*   e0     = (const int*)d_in[4];
    const int*   e1     = (const int*)d_in[5];
    const int*   e2     = (const int*)d_in[6];
    const float* w_lin1 = (const float*)d_in[7];
    const float* b_lin1 = (const float*)d_in[8];
    const float* mw0    = (const float*)d_in[9];
    const float* mb0    = (const float*)d_in[10];
    const float* mw1    = (const float*)d_in[11];
    const float* mb1    = (const float*)d_in[12];
    const float* wL0    = (const float*)d_in[13];
    const float* bL0    = (const float*)d_in[14];
    const float* wL1    = (const float*)d_in[15];
    const float* bL1    = (const float*)d_in[16];

    const int N0 = in_sizes[0] / LC;
    const int N1 = in_sizes[2] / 3;
    const int N2 = in_sizes[3] / 3;
    const int E0 = in_sizes[4] / 2;
    const int E1 = in_sizes[5] / 2;
    const int E2 = in_sizes[6] / 2;

    // Workspace: two ping-pong feature buffers [N2, 128] + kNN idx/weight.
    size_t S = (size_t)N2 * H;                 // floats per buffer
    float* bufA  = (float*)d_ws;
    float* bufB  = bufA + S;
    int*   knn_i = (int*)(bufB + S);
    float* knn_w = (float*)(knn_i + (size_t)N2 * KNN_K);

    float* out = (float*)d_out;

    // 1) x = elu(latent @ w_lin1 + b)  -> bufA [N0]
    {
        int tot = N0 * H;
        lin1_elu_kernel<<<(tot + 255) / 256, 256, 0, stream>>>(
            latent, w_lin1, b_lin1, bufA, N0);
    }

    // 2) two GIN convs on graph 0
    run_gin(bufA, bufB, e0, E0, N0, mw0 + 0 * H * H, mb0 + 0 * H,
            mw1 + 0 * H * H, mb1 + 0 * H, 1, stream);        // -> bufB
    run_gin(bufB, bufA, e0, E0, N0, mw0 + 1 * H * H, mb0 + 1 * H,
            mw1 + 1 * H * H, mb1 + 1 * H, 1, stream);        // -> bufA

    // 3) knn_interpolate level0 -> level1 (k = E0/N0 = 16)
    knn_topk_kernel<<<(N1 + KNN_BLOCK - 1) / KNN_BLOCK, KNN_BLOCK, 0, stream>>>(
        pos0, pos1, N0, N1, knn_i, knn_w);
    knn_gather_kernel<<<(N1 + 3) / 4, 128, 0, stream>>>(
        bufA, knn_i, knn_w, bufB, N1);                       // -> bufB [N1]

    // 4) GIN conv on graph 1
    run_gin(bufB, bufA, e1, E1, N1, mw0 + 2 * H * H, mb0 + 2 * H,
            mw1 + 2 * H * H, mb1 + 2 * H, 1, stream);        // -> bufA

    // 5) knn_interpolate level1 -> level2 (k = E1/N1 = 16)
    knn_topk_kernel<<<(N2 + KNN_BLOCK - 1) / KNN_BLOCK, KNN_BLOCK, 0, stream>>>(
        pos1, pos2, N1, N2, knn_i, knn_w);
    knn_gather_kernel<<<(N2 + 3) / 4, 128, 0, stream>>>(
        bufA, knn_i, knn_w, bufB, N2);                       // -> bufB [N2]

    // 6) GIN conv on graph 2
    run_gin(bufB, bufA, e2, E2, N2, mw0 + 3 * H * H, mb0 + 3 * H,
            mw1 + 3 * H * H, mb1 + 3 * H, 1, stream);        // -> bufA

    // 7) final GIN conv (hidden -> hidden -> 3), no outer ELU
    {
        long n4 = (long)N2 * H / 4;
        copy_f4_kernel<<<(unsigned)((n4 + 255) / 256), 256, 0, stream>>>(
            bufA, bufB, n4);
        long tw = (long)E2 * (H / 4);
        edge_agg_kernel<<<(unsigned)((tw + 255) / 256), 256, 0, stream>>>(
            e2, e2 + E2, bufA, bufB, E2);
        int gb = (N2 + 127) / 128;
        gemm_h128_bias_act_kernel<<<gb, 256, 0, stream>>>(bufB, wL0, bL0, N2, 1);
        gemm_out3_kernel<<<(N2 + 255) / 256, 256, 0, stream>>>(
            bufB, wL1, bL1, out, N2);
    }
}